// EuclideanCodebook_58428735094923
// MI455X (gfx1250) — compile-verified
//
#include <hip/hip_runtime.h>
#include <hip/hip_bf16.h>

// Problem constants (from reference): H=8, B=8, N=2048, D=64, C=1024
#define Hh 8
#define NT 16384           // B*N rows per head
#define Dd 64
#define Cc 1024
#define DECAY 0.8f
#define EPS 1e-5f

typedef float v2f __attribute__((ext_vector_type(2)));
typedef float v4f __attribute__((ext_vector_type(4)));
typedef float v8f __attribute__((ext_vector_type(8)));

// ---- output layout (floats, concatenated in reference return order) ----
static constexpr size_t OFF_QUANT = 0;                       // [H,NT,D]   8388608
static constexpr size_t OFF_IND   = 8388608;                 // [H,NT]      131072
static constexpr size_t OFF_OH    = 8519680;                 // [H,NT,C] 134217728
static constexpr size_t OFF_DIST  = 142737408;               // [H,NT,C] 134217728
static constexpr size_t OFF_ENORM = 276955136;               // [H,C,D]     524288
static constexpr size_t OFF_EA    = 277479424;               // [H,C,D]     524288
static constexpr size_t OFF_CS    = 278003712;               // [H,C]         8192

// ---------------------------------------------------------------- zero ws
__global__ void vq_zero_ws(float* ws, int n) {
  int i = blockIdx.x * blockDim.x + threadIdx.x;
  if (i < n) ws[i] = 0.0f;
}

// ------------------------------------------------- code squared-norm table
// one thread per (h,c): esq[h*C+c] = sum_d embed[h,c,d]^2   (8192 threads)
__global__ void vq_esq(const float* __restrict__ embed, float* __restrict__ esq) {
  int i = blockIdx.x * blockDim.x + threadIdx.x;     // i = h*C + c
  if (i >= Hh * Cc) return;
  const float* e = embed + (size_t)i * Dd;
  float s = 0.0f;
#pragma unroll
  for (int k = 0; k < Dd / 4; ++k) {
    v4f v = *(const v4f*)(e + 4 * k);
    s += v.x * v.x + v.y * v.y + v.z * v.z + v.w * v.w;
  }
  esq[i] = s;
}

// ---------------------------------------------------------- main fused kernel
// grid: (NT/16, H), block: 256 (8 waves). Wave w handles codes [w*128, w*128+128).
__global__ void __launch_bounds__(256)
vq_dist_argmax(const float* __restrict__ x, const float* __restrict__ embed,
               const float* __restrict__ esq_tab, float* __restrict__ out,
               float* __restrict__ esum, float* __restrict__ counts) {
  const int tid  = threadIdx.x;
  const int wid  = tid >> 5;
  const int lane = tid & 31;
  const int half = lane >> 4;     // 0: K%4 in {0,1}   1: K%4 in {2,3}
  const int l16  = lane & 15;
  const int h    = blockIdx.y;
  const int m0   = blockIdx.x * 16;

  float* quant_out = out + OFF_QUANT;
  float* ind_out   = out + OFF_IND;
  float* oh_out    = out + OFF_OH;
  float* dist_out  = out + OFF_DIST;

  // ---- load A fragments (16x64 fp32 tile, WMMA A layout), fold ||f||^2 ----
  const float* xrow = x + ((size_t)h * NT + (size_t)(m0 + l16)) * Dd + 2 * half;
  v2f afrag[16];
  float fpart = 0.0f;
#pragma unroll
  for (int k = 0; k < 16; ++k) {
    v2f a = *(const v2f*)(xrow + 4 * k);
    afrag[k] = a;
    fpart += a.x * a.x + a.y * a.y;
  }
  // combine lane halves -> every lane holds ||f||^2 of row (lane&15)
  float fsq_full = fpart + __shfl_xor(fpart, 16, 32);
  float fsq[8];
#pragma unroll
  for (int j = 0; j < 8; ++j)
    fsq[j] = __shfl(fsq_full, j + 8 * half, 32);   // row for acc VGPR j in this half

  // ---- per-row running argmax over this wave's 128 codes ----
  float bv[8]; int bi[8];
#pragma unroll
  for (int j = 0; j < 8; ++j) { bv[j] = -3.4e38f; bi[j] = 0; }

  const size_t rowbaseC = ((size_t)h * NT + m0) * (size_t)Cc;

  // B fragments double-buffered: tile t covers codes [wid*128 + t*16, +16)
  const float* erowbase =
      embed + ((size_t)h * Cc + (size_t)(wid * 128 + l16)) * Dd + 2 * half;
  const float* esqbase = esq_tab + (size_t)h * Cc + (wid * 128 + l16);

  v2f bbuf0[16], bbuf1[16];
  float esq0, esq1;
#pragma unroll
  for (int k = 0; k < 16; ++k)
    bbuf0[k] = *(const v2f*)(erowbase + 4 * k);       // prefetch tile 0
  esq0 = esqbase[0];

#pragma unroll
  for (int t = 0; t < 8; ++t) {
    v2f* cur = (t & 1) ? bbuf1 : bbuf0;
    v2f* nxt = (t & 1) ? bbuf0 : bbuf1;
    float esq = (t & 1) ? esq1 : esq0;
    float* esqn = (t & 1) ? &esq0 : &esq1;

    // issue next tile's independent loads before consuming current tile
    if (t < 7) {
      const float* pn = erowbase + (size_t)(t + 1) * 16 * Dd;
#pragma unroll
      for (int k = 0; k < 16; ++k)
        nxt[k] = *(const v2f*)(pn + 4 * k);
      *esqn = esqbase[(t + 1) * 16];
    }

    const int c0 = wid * 128 + t * 16;
    v8f acc = {0.f, 0.f, 0.f, 0.f, 0.f, 0.f, 0.f, 0.f};
#pragma unroll
    for (int k = 0; k < 16; ++k) {
      acc = __builtin_amdgcn_wmma_f32_16x16x4_f32(
          /*neg_a=*/false, afrag[k], /*neg_b=*/false, cur[k],
          /*c_mod=*/(short)0, acc, /*reuse_a=*/false, /*reuse_b=*/false);
    }
    const int cc = c0 + l16;
#pragma unroll
    for (int j = 0; j < 8; ++j) {
      const int rl = j + 8 * half;                    // local row of acc[j]
      float dv = 2.0f * acc[j] - fsq[j] - esq;        // -(||f||^2 - 2fe + ||e||^2)
      size_t off = rowbaseC + (size_t)rl * Cc + cc;
      dist_out[off] = dv;
      oh_out[off]   = 0.0f;                           // zero-fill onehot
      if (dv > bv[j] || (dv == bv[j] && cc < bi[j])) { bv[j] = dv; bi[j] = cc; }
    }
  }

  // ---- reduce argmax across the 16 lanes of each half (same row set) ----
#pragma unroll
  for (int j = 0; j < 8; ++j) {
#pragma unroll
    for (int off = 8; off >= 1; off >>= 1) {
      float ov = __shfl_xor(bv[j], off, 32);
      int   oi = __shfl_xor(bi[j], off, 32);
      if (ov > bv[j] || (ov == bv[j] && oi < bi[j])) { bv[j] = ov; bi[j] = oi; }
    }
  }

  __shared__ float s_val[8][16];
  __shared__ int   s_id[8][16];
  __shared__ int   s_fid[16];
  if (l16 == 0) {
#pragma unroll
    for (int j = 0; j < 8; ++j) {
      s_val[wid][j + 8 * half] = bv[j];
      s_id[wid][j + 8 * half]  = bi[j];
    }
  }
  // drain this wave's global zero-stores before the 1.0 overwrite after barrier
  asm volatile("s_wait_storecnt 0" ::: "memory");
  __syncthreads();

  if (tid < 16) {
    float fv = s_val[0][tid]; int fi = s_id[0][tid];
#pragma unroll
    for (int w = 1; w < 8; ++w) {
      float vv = s_val[w][tid]; int ii = s_id[w][tid];
      if (vv > fv || (vv == fv && ii < fi)) { fv = vv; fi = ii; }
    }
    s_fid[tid] = fi;
    const int m = m0 + tid;
    ind_out[(size_t)h * NT + m] = (float)fi;
    oh_out[((size_t)h * NT + m) * (size_t)Cc + fi] = 1.0f;
    atomicAdd(counts + (size_t)h * Cc + fi, 1.0f);
  }
  __syncthreads();

  // ---- quantize gather + embed_sum scatter-add (16 rows x 64 dims) ----
  for (int i = tid; i < 16 * Dd; i += 256) {
    const int r = i >> 6, dd = i & 63;
    const int idx = s_fid[r];
    const int m = m0 + r;
    float ev = embed[((size_t)h * Cc + idx) * Dd + dd];
    quant_out[((size_t)h * NT + m) * Dd + dd] = ev;
    float xv = x[((size_t)h * NT + m) * Dd + dd];
    atomicAdd(esum + ((size_t)h * Cc + idx) * Dd + dd, xv);
  }
}

// ------------------------------------------------------------- EMA finalize
// one block (256 threads) per head
__global__ void __launch_bounds__(256)
vq_finalize(const float* __restrict__ embed_avg, const float* __restrict__ cluster_size,
            const float* __restrict__ esum, const float* __restrict__ counts,
            float* __restrict__ out) {
  const int h = blockIdx.x;
  const int tid = threadIdx.x;
  __shared__ float sred[256];

  float* cs_out = out + OFF_CS;
  float* ea_out = out + OFF_EA;
  float* en_out = out + OFF_ENORM;

  float csn[4];
  float local = 0.0f;
#pragma unroll
  for (int q = 0; q < 4; ++q) {
    const int c = tid + q * 256;
    float cs = cluster_size[(size_t)h * Cc + c] * DECAY +
               counts[(size_t)h * Cc + c] * (1.0f - DECAY);
    csn[q] = cs; local += cs;
    cs_out[(size_t)h * Cc + c] = cs;
  }
  sred[tid] = local;
  __syncthreads();
  for (int s = 128; s > 0; s >>= 1) {
    if (tid < s) sred[tid] += sred[tid + s];
    __syncthreads();
  }
  const float ntot = sred[0];

#pragma unroll
  for (int q = 0; q < 4; ++q) {
    const int c = tid + q * 256;
    const float smooth = (csn[q] + EPS) / (ntot + Cc * EPS) * ntot;
    const float inv = 1.0f / smooth;
    for (int dd = 0; dd < Dd; ++dd) {
      const size_t o = ((size_t)h * Cc + c) * Dd + dd;
      float ea = embed_avg[o] * DECAY + esum[o] * (1.0f - DECAY);
      ea_out[o] = ea;
      en_out[o] = ea * inv;
    }
  }
}

extern "C" void kernel_launch(void* const* d_in, const int* in_sizes, int n_in,
                              void* d_out, int out_size, void* d_ws, size_t ws_size,
                              hipStream_t stream) {
  const float* x            = (const float*)d_in[0];  // [H,B,N,D]
  const float* embed        = (const float*)d_in[1];  // [H,C,D]
  const float* embed_avg    = (const float*)d_in[2];  // [H,C,D]
  const float* cluster_size = (const float*)d_in[3];  // [H,C]
  float* out = (float*)d_out;

  float* ws     = (float*)d_ws;
  float* esum   = ws;                                  // [H,C,D]
  float* counts = ws + (size_t)Hh * Cc * Dd;           // [H,C]
  float* esq    = counts + (size_t)Hh * Cc;            // [H,C]

  const int nz = Hh * Cc * Dd + Hh * Cc;
  vq_zero_ws<<<(nz + 255) / 256, 256, 0, stream>>>(ws, nz);
  vq_esq<<<(Hh * Cc + 255) / 256, 256, 0, stream>>>(embed, esq);

  dim3 grid(NT / 16, Hh);
  vq_dist_argmax<<<grid, 256, 0, stream>>>(x, embed, esq, out, esum, counts);

  vq_finalize<<<Hh, 256, 0, stream>>>(embed_avg, cluster_size, esum, counts, out);
}